// QuixerTimeSeries_22866405884517
// MI455X (gfx1250) — compile-verified
//
#include <hip/hip_runtime.h>
#include <hip/hip_bf16.h>
#include <math.h>

typedef __attribute__((ext_vector_type(2))) float v2f;
typedef __attribute__((ext_vector_type(8))) float v8f;

#define NQ    6
#define DIMQ  64
#define NT    32
#define NROTS 48
#define BATCHN 1024
#define FEATN  64
#define OUTD   16

// sim14 gate pattern for one layer (24 gates). kind 0=RY(w0), 1=CRX(ctrl=w0,tgt=w1)
namespace {
constexpr int GK[24]  = {0,0,0,0,0,0, 1,1,1,1,1,1, 0,0,0,0,0,0, 1,1,1,1,1,1};
constexpr int GW0[24] = {0,1,2,3,4,5, 5,4,3,2,1,0, 0,1,2,3,4,5, 5,0,1,2,3,4};
constexpr int GW1[24] = {0,0,0,0,0,0, 0,5,4,3,2,1, 0,0,0,0,0,0, 4,5,0,1,2,3};
}

// Apply NG gates of the sim14 circuit to a wave-private 64-amplitude LDS state.
// Wire w corresponds to amplitude-index bit (5 - w).
template<int NG>
__device__ __forceinline__ void apply_sim14(float2* st, const float* __restrict__ thetas, int lane) {
#pragma unroll
  for (int g = 0; g < NG; ++g) {
    const int gl = g % 24;
    float sn, cs;
    __sincosf(0.5f * thetas[g], &sn, &cs);
    if (GK[gl] == 0) {
      // RY on wire GW0: 32 pairs, one per lane
      const int bp   = 5 - GW0[gl];
      const int mask = 1 << bp;
      const int i0 = ((lane & ~(mask - 1)) << 1) | (lane & (mask - 1));
      const int i1 = i0 | mask;
      float2 a0 = st[i0], a1 = st[i1];
      st[i0] = make_float2(cs * a0.x - sn * a1.x, cs * a0.y - sn * a1.y);
      st[i1] = make_float2(sn * a0.x + cs * a1.x, sn * a0.y + cs * a1.y);
    } else {
      // CRX: only ctrl-bit==1 half of the state; 16 pairs on lanes 0..15
      if (lane < 16) {
        const int pc = 5 - GW0[gl], pt = 5 - GW1[gl];
        const int p1 = pc < pt ? pc : pt;
        const int p2 = pc < pt ? pt : pc;
        const int b1 = (p1 == pc) ? 1 : 0;
        const int b2 = (p2 == pc) ? 1 : 0;
        int v = lane;
        v = ((v >> p1) << (p1 + 1)) | (b1 << p1) | (v & ((1 << p1) - 1));
        v = ((v >> p2) << (p2 + 1)) | (b2 << p2) | (v & ((1 << p2) - 1));
        const int i0 = v;              // ctrl=1, tgt=0
        const int i1 = v | (1 << pt);  // ctrl=1, tgt=1
        float2 t0 = st[i0], t1 = st[i1];
        // new0 = c*t0 - i*s*t1 ; new1 = -i*s*t0 + c*t1
        st[i0] = make_float2(cs * t0.x + sn * t1.y, cs * t0.y - sn * t1.x);
        st[i1] = make_float2(cs * t1.x + sn * t0.y, cs * t1.y - sn * t0.x);
      }
    }
    __builtin_amdgcn_wave_barrier();  // cross-lane LDS deps invisible to compiler
  }
}

// ---------------- Kernel 1: P = sigmoid(x @ fp_w^T + fp_b) via fp32 WMMA ----
// One wave = one 16x16 tile of the 32768x48 output. K=64 -> 16 x v_wmma_f32_16x16x4_f32.
__global__ void __launch_bounds__(256)
k_fgemm_sigmoid(const float* __restrict__ x, const float* __restrict__ w,
                const float* __restrict__ bias, float* __restrict__ P,
                float* __restrict__ meanSlot) {
  const int tid  = threadIdx.x;
  const int lane = tid & 31;
  const int wav  = tid >> 5;
  const int tile = blockIdx.x * 8 + wav;   // 768 blocks * 8 waves = 6144 tiles
  const int tn = tile % 3;                 // 48 cols = 3 tiles
  const int tm = tile / 3;                 // 32768 rows = 2048 tiles
  const int ln = lane & 15;
  const int hi = lane >> 4;                // lanes 16..31 hold K+2,K+3

  const float* arow = x + (size_t)(tm * 16 + ln) * FEATN + 2 * hi;  // A[m][k]
  const float* brow = w + (size_t)(tn * 16 + ln) * FEATN + 2 * hi;  // B[k][n] = fp_w[n][k]
  v8f c = {0.f, 0.f, 0.f, 0.f, 0.f, 0.f, 0.f, 0.f};
#pragma unroll
  for (int k0 = 0; k0 < FEATN; k0 += 4) {
    v2f a; a.x = arow[k0]; a.y = arow[k0 + 1];
    v2f b; b.x = brow[k0]; b.y = brow[k0 + 1];
    c = __builtin_amdgcn_wmma_f32_16x16x4_f32(false, a, false, b, (short)0, c, false, false);
  }
  const float bv = bias[tn * 16 + ln];
#pragma unroll
  for (int r = 0; r < 8; ++r) {
    const int m = tm * 16 + r + hi * 8;    // C/D layout: VGPR r = row r (+8 for hi lanes)
    float v = c[r] + bv;
    v = 1.f / (1.f + __expf(-v));
    P[(size_t)m * NROTS + tn * 16 + ln] = v;
  }
  if (blockIdx.x == 0 && tid == 0) meanSlot[0] = 0.f;  // zero mean accumulator
}

// ---------------- Kernel 2: fused quantum pipeline, one workgroup per batch --
__global__ void __launch_bounds__(1024)
k_quixer(const float* __restrict__ P,
         const float* __restrict__ poly,
         const float* __restrict__ mix_re,
         const float* __restrict__ mix_im,
         const float* __restrict__ qff,
         const float* __restrict__ out_w,
         const float* __restrict__ out_b,
         float* __restrict__ out) {
  __shared__ float2 sSt[NT][DIMQ];   // per-timestep states (16 KB)
  __shared__ float2 sWork[DIMQ];
  __shared__ float2 sAcc[DIMQ];
  __shared__ float  sExps[3 * NQ];
  __shared__ float  sScal[2];

  const int b    = blockIdx.x;
  const int tid  = threadIdx.x;
  const int lane = tid & 31;
  const int wav  = tid >> 5;         // timestep handled by this wave

  if (tid < DIMQ) {
    sWork[tid] = make_float2(tid == 0 ? 1.f : 0.f, 0.f);               // base = e0
    sAcc[tid]  = make_float2(tid == 0 ? poly[0] : 0.f, 0.f);           // acc = c0*base
  }
  __syncthreads();

  const float* Pb = P + ((size_t)b * NT + wav) * NROTS;
  for (int k = 1; k <= 3; ++k) {
    // each wave copies the shared work-state into its private region
    sSt[wav][lane]      = sWork[lane];
    sSt[wav][lane + 32] = sWork[lane + 32];
    __builtin_amdgcn_wave_barrier();
    apply_sim14<48>(&sSt[wav][0], Pb, lane);   // encoder, 2 layers
    __syncthreads();
    if (tid < DIMQ) {                          // work = sum_t lcu[t]*st[t]
      float sr = 0.f, si = 0.f;
#pragma unroll
      for (int t = 0; t < NT; ++t) {
        const float2 v = sSt[t][tid];
        const float lr = mix_re[t], li = mix_im[t];
        sr += lr * v.x - li * v.y;
        si += lr * v.y + li * v.x;
      }
      sWork[tid] = make_float2(sr, si);
      const float pk = poly[k];
      sAcc[tid].x += pk * sr;
      sAcc[tid].y += pk * si;
    }
    __syncthreads();
  }

  // mixed = acc / sum|coeffs|
  if (tid < DIMQ) {
    const float sa  = fabsf(poly[0]) + fabsf(poly[1]) + fabsf(poly[2]) + fabsf(poly[3]);
    const float inv = 1.f / sa;
    sAcc[tid].x *= inv;
    sAcc[tid].y *= inv;
  }
  __syncthreads();
  if (tid == 0) {                              // final_prob = ||mixed||_2
    float n2 = 0.f;
    for (int i = 0; i < DIMQ; ++i) n2 += sAcc[i].x * sAcc[i].x + sAcc[i].y * sAcc[i].y;
    const float fp = sqrtf(n2);
    sScal[0] = 1.f / fmaxf(fp, 1e-12f);
    sScal[1] = fp;
  }
  __syncthreads();
  if (tid < DIMQ) {
    sWork[tid].x = sAcc[tid].x * sScal[0];
    sWork[tid].y = sAcc[tid].y * sScal[0];
  }
  __syncthreads();
  if (tid < 32) apply_sim14<24>(&sWork[0], qff, lane);   // QFF, 1 layer, wave 0
  __syncthreads();

  // measure_xyz: thread w handles wire w
  if (tid < NQ) {
    const int bp   = 5 - tid;
    const int mask = 1 << bp;
    float cr = 0.f, ci = 0.f, ez = 0.f;
    for (int p = 0; p < 32; ++p) {
      const int i0 = ((p & ~(mask - 1)) << 1) | (p & (mask - 1));
      const int i1 = i0 | mask;
      const float2 s0 = sWork[i0], s1 = sWork[i1];
      cr += s0.x * s1.x + s0.y * s1.y;           // Re(conj(s0)*s1)
      ci += s0.x * s1.y - s0.y * s1.x;           // Im(conj(s0)*s1)
      ez += s0.x * s0.x + s0.y * s0.y - s1.x * s1.x - s1.y * s1.y;
    }
    sExps[tid]          = 2.f * cr;
    sExps[NQ + tid]     = 2.f * ci;
    sExps[2 * NQ + tid] = ez;
  }
  __syncthreads();

  if (tid < OUTD) {                              // op = exps @ out_w^T + out_b
    float acc = out_b[tid];
#pragma unroll
    for (int j = 0; j < 3 * NQ; ++j) acc += sExps[j] * out_w[tid * (3 * NQ) + j];
    out[(size_t)b * OUTD + tid] = acc;
  }
  if (tid == 0) atomicAdd(out + BATCHN * OUTD, sScal[1] * (1.f / BATCHN));
}

extern "C" void kernel_launch(void* const* d_in, const int* in_sizes, int n_in,
                              void* d_out, int out_size, void* d_ws, size_t ws_size,
                              hipStream_t stream) {
  const float* x      = (const float*)d_in[0];  // (1024,32,64)
  const float* fp_w   = (const float*)d_in[1];  // (48,64)
  const float* fp_b   = (const float*)d_in[2];  // (48,)
  const float* poly   = (const float*)d_in[3];  // (4,)
  const float* mix_re = (const float*)d_in[4];  // (32,)
  const float* mix_im = (const float*)d_in[5];  // (32,)
  const float* qff    = (const float*)d_in[6];  // (48,) -> first 24 used
  const float* out_w  = (const float*)d_in[7];  // (16,18)
  const float* out_b  = (const float*)d_in[8];  // (16,)
  float* out = (float*)d_out;                   // 1024*16 op + 1 mean
  float* P   = (float*)d_ws;                    // (32768,48) sigmoid params, 6.3 MB

  // 2048 M-tiles * 3 N-tiles = 6144 waves; 8 waves (256 thr) per block
  k_fgemm_sigmoid<<<768, 256, 0, stream>>>(x, fp_w, fp_b, P, out + BATCHN * OUTD);
  k_quixer<<<BATCHN, 1024, 0, stream>>>(P, poly, mix_re, mix_im, qff, out_w, out_b, out);
}